// Floss_36335423324332
// MI455X (gfx1250) — compile-verified
//
#include <hip/hip_runtime.h>
#include <math.h>
#include <stdint.h>

#define TPB      256
#define WIDTH    4096
#define VPT      (WIDTH / TPB)   // 16 bins per thread
#define POS_NUM  10
#define NEG_NUM  30              // min(3*POS_NUM, W-1)
#define LAMBDA_FOV 1.0f
#define PI_F 3.14159265358979323846f
#define BPW  5                   // batches per wave in prep (5 * 3 rows = 15 <= 16)

typedef __attribute__((ext_vector_type(2))) float v2f;
typedef __attribute__((ext_vector_type(8))) float v8f;

// ---------- small 3x3 helpers ----------
__device__ __forceinline__ void inv3(const float m[9], float o[9]) {
  float a=m[0],b=m[1],c=m[2],d=m[3],e=m[4],f=m[5],g=m[6],h=m[7],i=m[8];
  float A  =  (e*i - f*h);
  float Bc = -(d*i - f*g);
  float Cc =  (d*h - e*g);
  float det = a*A + b*Bc + c*Cc;
  float r = 1.0f / det;
  o[0]=A*r;          o[1]=(c*h-b*i)*r;  o[2]=(b*f-c*e)*r;
  o[3]=Bc*r;         o[4]=(a*i-c*g)*r;  o[5]=(c*d-a*f)*r;
  o[6]=Cc*r;         o[7]=(b*g-a*h)*r;  o[8]=(a*e-b*d)*r;
}

// ---------- kernel 1: one wave per 5 batches; 3x3 products via V_WMMA_F32_16X16X4_F32 ----------
// A(16x4): rows 3b+i hold E^b[i][0..2] (col 3 = 0).  B(4x16): cols 3b+j hold Rinv^b[0..2][j]
// (row 3 = 0).  D = A@B: diagonal 3x3 blocks = E^b @ Rinv^b.
__global__ __launch_bounds__(32) void fov_prep_kernel(
    const float* __restrict__ gtT,
    const float* __restrict__ gtE,
    const float* __restrict__ prE,
    float* __restrict__ out_fl,   // [B,4,4]
    int*   __restrict__ xmin_out, // [B]
    float* __restrict__ accum,    // [2] num, den
    int Bn, int Wn)
{
  __shared__ float sRinv[BPW * 9];
  __shared__ float sD[16 * 16];

  const int L    = threadIdx.x;          // 0..31
  const int base = blockIdx.x * BPW;

  if (blockIdx.x == 0 && L == 0) { accum[0] = 0.0f; accum[1] = 0.0f; }

  // --- leaders (lanes 0..4): Rinv, yaw -> xmin ---
  float rinv[9];
  const bool leader = (L < BPW) && (base + L < Bn);
  if (leader) {
    const int bidx = base + L;
    float g[9];
    #pragma unroll
    for (int i = 0; i < 3; ++i)
      #pragma unroll
      for (int j = 0; j < 3; ++j)
        g[i*3+j] = gtT[(size_t)bidx*16 + i*4 + j];
    inv3(g, rinv);
    #pragma unroll
    for (int k = 0; k < 9; ++k) sRinv[L*9 + k] = rinv[k];

    // axis = (pred_R @ Rinv) e1 -> first column; only rows 0,1 of pred_R needed
    const float* pr = prE + (size_t)bidx * 16;
    float c0 = rinv[0], c1 = rinv[3], c2 = rinv[6];
    float ax = pr[0]*c0 + pr[1]*c1 + pr[2]*c2;
    float ay = pr[4]*c0 + pr[5]*c1 + pr[6]*c2;
    float yaw  = atan2f(ay, ax);
    float fidx = (-yaw + PI_F) * (float)Wn / (2.0f * PI_F);
    xmin_out[bidx] = (int)fidx - POS_NUM / 2;   // trunc == floor since fidx >= 0
  }
  __syncthreads();

  // --- build WMMA fragments (all 32 lanes; EXEC all-1s at the WMMA) ---
  const int half = L >> 4;        // K-half: 0 -> K={0,1}, 1 -> K={2,3}
  const int r    = L & 15;        // A row index / B column index
  const int rb   = r / 3;         // which 3x3 block
  const int ri   = r - 3 * rb;    // row/col within block
  const bool bvalid = (r < 15) && (base + rb < Bn);

  v2f afrag = {0.0f, 0.0f};
  if (bvalid) {
    const float* erow = gtE + (size_t)(base + rb) * 16 + ri * 4;
    if (half == 0) { afrag.x = erow[0]; afrag.y = erow[1]; }   // K=0,1
    else           { afrag.x = erow[2]; afrag.y = 0.0f;    }   // K=2,3 (pad)
  }

  v2f bfrag = {0.0f, 0.0f};
  if (bvalid) {
    const float* rv = &sRinv[rb * 9];
    if (half == 0) { bfrag.x = rv[0*3 + ri]; bfrag.y = rv[1*3 + ri]; } // rows K=0,1
    else           { bfrag.x = rv[2*3 + ri]; bfrag.y = 0.0f;         } // rows K=2,3
  }

  v8f cz = {};
  v8f d = __builtin_amdgcn_wmma_f32_16x16x4_f32(
      /*neg_a=*/false, afrag, /*neg_b=*/false, bfrag,
      /*c_mod=*/(short)0, cz, /*reuse_a=*/false, /*reuse_b=*/false);

  // spill D to LDS: VGPR rr holds row rr (lanes 0-15) / row rr+8 (lanes 16-31)
  #pragma unroll
  for (int rr = 0; rr < 8; ++rr)
    sD[(half * 8 + rr) * 16 + r] = d[rr];
  __syncthreads();

  // --- leaders: invert diagonal block, write f_l = eye(4) with [:3,:3] = inv(E@Rinv) ---
  if (leader) {
    const int bidx = base + L;
    float fg[9], r33[9];
    #pragma unroll
    for (int i = 0; i < 3; ++i)
      #pragma unroll
      for (int j = 0; j < 3; ++j)
        fg[i*3+j] = sD[(3*L + i) * 16 + (3*L + j)];
    inv3(fg, r33);
    float* o = out_fl + (size_t)bidx * 16;
    #pragma unroll
    for (int i = 0; i < 3; ++i) {
      o[i*4+0] = r33[i*3+0]; o[i*4+1] = r33[i*3+1]; o[i*4+2] = r33[i*3+2]; o[i*4+3] = 0.0f;
    }
    o[12] = 0.0f; o[13] = 0.0f; o[14] = 0.0f; o[15] = 1.0f;
  }
}

// ---------- kernel 2: one block per row; async-stage to LDS, BCE, top-30 ----------
__global__ __launch_bounds__(TPB) void fov_row_kernel(
    const float* __restrict__ pred,     // [B,W]
    const int*   __restrict__ xmin_arr, // [B]
    float* __restrict__ out_gt,         // [B,W]
    float* __restrict__ accum)          // [2]
{
  __shared__ float srow[WIDTH];                       // 16 KB staged row
  __shared__ unsigned long long swmax[TPB / 32];
  __shared__ unsigned long long swin;
  __shared__ float spos[TPB / 32];
  __shared__ int soverlap;

  const int t = threadIdx.x;
  const int b = blockIdx.x;
  const float* rowp = pred + (size_t)b * WIDTH;

  // Async DMA the whole row into LDS: 4 x b128 per thread (256 thr * 16B * 4)
  #pragma unroll
  for (int m = 0; m < 4; ++m) {
    int idx = (m * TPB + t) * 4;                      // float index, 16B aligned
    unsigned lo = (unsigned)(size_t)(&srow[idx]);     // low 32 bits = LDS offset
    const float* ga = rowp + idx;
    asm volatile("global_load_async_to_lds_b128 %0, %1, off"
                 :: "v"(lo), "v"(ga) : "memory");
  }
  if (t == 0) soverlap = 0;
  asm volatile("s_wait_asynccnt 0x0" ::: "memory");
  __syncthreads();

  const int xm = xmin_arr[b];

  // key = (bits(loss_c) << 32) | (W-1-c): max picks largest value, ties -> lowest c
  unsigned long long key[VPT];
  float posSum = 0.0f;
  #pragma unroll
  for (int k = 0; k < VPT; ++k) {
    int c = t + k * TPB;
    float pv = srow[c];
    int d = c - xm;
    if (d < 0) d += WIDTH;
    if (d >= WIDTH) d -= WIDTH;
    bool pos = (d < POS_NUM);
    out_gt[(size_t)b * WIDTH + c] = pos ? 1.0f : 0.0f;
    float bce = pos ? -fmaxf(logf(pv),    -100.0f)
                    : -fmaxf(log1pf(-pv), -100.0f);
    float lc;
    if (pos) { posSum += bce; lc = 0.0f; } else { lc = bce; }
    key[k] = ((unsigned long long)__float_as_uint(lc) << 32)
           | (unsigned)(WIDTH - 1 - c);
  }

  unsigned rem = 0;                     // per-thread removed mask
  unsigned long long best = 0ULL;
  #pragma unroll
  for (int k = 0; k < VPT; ++k) best = key[k] > best ? key[k] : best;

  float negSum = 0.0f;                  // accumulated on t==0 only
  for (int it = 0; it < NEG_NUM; ++it) {
    // wave32 max reduction of `best`
    unsigned long long w = best;
    #pragma unroll
    for (int off = 16; off > 0; off >>= 1) {
      unsigned long long o = __shfl_xor(w, off, 32);
      w = o > w ? o : w;
    }
    if ((t & 31) == 0) swmax[t >> 5] = w;
    __syncthreads();
    if (t == 0) {
      unsigned long long m = swmax[0];
      #pragma unroll
      for (int i = 1; i < TPB / 32; ++i) m = swmax[i] > m ? swmax[i] : m;
      swin = m;
      negSum += __uint_as_float((unsigned)(m >> 32));
    }
    __syncthreads();
    unsigned long long win = swin;
    if (win != 0ULL) {
      int wc = WIDTH - 1 - (int)(win & 0xFFFFFFFFu);
      if ((wc & (TPB - 1)) == t) {      // unique owner: column index identifies it
        int k = wc >> 8;                // wc / TPB
        rem |= 1u << k;
        int d = wc - xm;
        if (d < 0) d += WIDTH;
        if (d >= WIDTH) d -= WIDTH;
        if (d < POS_NUM) atomicAdd(&soverlap, 1);   // degenerate pos-in-top30 case
        best = 0ULL;                    // rebuild only this thread's best
        #pragma unroll
        for (int k2 = 0; k2 < VPT; ++k2)
          if (!(rem & (1u << k2)) && key[k2] > best) best = key[k2];
      }
    }
  }

  // block reduce posSum, then global atomic accumulate
  float ps = posSum;
  #pragma unroll
  for (int off = 16; off > 0; off >>= 1) ps += __shfl_xor(ps, off, 32);
  if ((t & 31) == 0) spos[t >> 5] = ps;
  __syncthreads();
  if (t == 0) {
    float tot = 0.0f;
    #pragma unroll
    for (int i = 0; i < TPB / 32; ++i) tot += spos[i];
    atomicAdd(&accum[0], tot + negSum);
    atomicAdd(&accum[1], (float)(POS_NUM + NEG_NUM - soverlap));
  }
}

// ---------- kernel 3: final scalar ----------
__global__ void fov_finalize_kernel(const float* __restrict__ accum,
                                    float* __restrict__ out)
{
  out[0] = LAMBDA_FOV * accum[0] / accum[1];
}

extern "C" void kernel_launch(void* const* d_in, const int* in_sizes, int n_in,
                              void* d_out, int out_size, void* d_ws, size_t ws_size,
                              hipStream_t stream) {
  const float* gtT  = (const float*)d_in[0];  // [B,4,4]
  const float* gtE  = (const float*)d_in[1];  // [B,4,4]
  const float* prE  = (const float*)d_in[2];  // [B,4,4]
  const float* pred = (const float*)d_in[3];  // [B,W]

  const int Bn = in_sizes[0] / 16;
  const int Wn = in_sizes[3] / Bn;            // == 4096 (compile-time constants assume this)

  float* out    = (float*)d_out;
  float* out_gt = out + 1;                            // [B,W]
  float* out_fl = out + 1 + (size_t)Bn * Wn;          // [B,4,4]

  int*   xm    = (int*)d_ws;
  float* accum = (float*)((char*)d_ws + (((size_t)Bn * sizeof(int) + 255) & ~(size_t)255));

  fov_prep_kernel<<<(Bn + BPW - 1) / BPW, 32, 0, stream>>>(
      gtT, gtE, prE, out_fl, xm, accum, Bn, Wn);
  fov_row_kernel<<<Bn, TPB, 0, stream>>>(pred, xm, out_gt, accum);
  fov_finalize_kernel<<<1, 1, 0, stream>>>(accum, out);
}